// UVTo3D_74689481278081
// MI455X (gfx1250) — compile-verified
//
#include <hip/hip_runtime.h>
#include <math.h>

// Problem constants (from reference)
#define BN    524288      // B*N points
#define NPB   32768       // N points per batch (2^15)
#define CH    128         // feature channels
#define NV    642         // vertices
#define NVP   656         // vertices padded to multiple of 16 (41*16)
#define NF    1280        // faces
#define UVW   1001

typedef __attribute__((ext_vector_type(2))) float v2f;
typedef __attribute__((ext_vector_type(8))) float v8f;

// ---- gfx1250 async global->LDS staging (ASYNCcnt path), with safe fallback --
#if defined(__has_builtin)
#if __has_builtin(__builtin_amdgcn_global_load_async_to_lds_b32)
#define HAVE_ASYNC_LDS 1
#endif
#endif
#ifndef HAVE_ASYNC_LDS
#define HAVE_ASYNC_LDS 0
#endif

#if HAVE_ASYNC_LDS
typedef __attribute__((address_space(1))) int* gas_i32p;   // global (AS1)
typedef __attribute__((address_space(3))) int* las_i32p;   // LDS (AS3)
#endif

template <typename T>
__device__ __forceinline__ void stage_to_lds(T* lds, const T* g, int n, int t,
                                             int nthreads) {
#if HAVE_ASYNC_LDS
  for (int idx = t; idx < n; idx += nthreads) {
    int* gp = (int*)(g + idx);    // generic int* (reinterpret + drop const)
    int* lp = (int*)(lds + idx);  // generic int*
    __builtin_amdgcn_global_load_async_to_lds_b32(
        (gas_i32p)gp, (las_i32p)lp, /*imm offset=*/0, /*cpol=*/0);
  }
#else
  for (int idx = t; idx < n; idx += nthreads) lds[idx] = g[idx];
#endif
}

__device__ __forceinline__ void wait_async_then_barrier() {
#if HAVE_ASYNC_LDS
#if __has_builtin(__builtin_amdgcn_s_wait_asynccnt)
  __builtin_amdgcn_s_wait_asynccnt(0);
#else
  asm volatile("s_wait_asynccnt 0x0" ::: "memory");
#endif
#endif
  __syncthreads();
}

__device__ __forceinline__ float3 mk3(float x, float y, float z) {
  float3 r; r.x = x; r.y = y; r.z = z; return r;
}
__device__ __forceinline__ float3 sub3(float3 a, float3 b) {
  return mk3(a.x - b.x, a.y - b.y, a.z - b.z);
}
__device__ __forceinline__ float3 cross3(float3 a, float3 b) {
  return mk3(a.y * b.z - a.z * b.y, a.z * b.x - a.x * b.z, a.x * b.y - a.y * b.x);
}
__device__ __forceinline__ float norm3(float3 a) {
  return sqrtf(a.x * a.x + a.y * a.y + a.z * a.z);
}
__device__ __forceinline__ float3 uvTo3d(float u, float v) {
  float phi = 6.28318530717958647692f * (u - 0.5f);
  float th  = 3.14159265358979323846f * (v - 0.5f);
  float sp, cp, st, ct;
  sincosf(phi, &sp, &cp);
  sincosf(th, &st, &ct);
  return mk3(ct * cp, ct * sp, st);
}

// ---------------------------------------------------------------------------
// Build face->vertex incidence matrix M (NVP x NF, row-major), entries 0..3.
// ---------------------------------------------------------------------------
__global__ void k_build_M(const int* __restrict__ faces, float* __restrict__ M) {
  int f = blockIdx.x * blockDim.x + threadIdx.x;
  if (f >= NF) return;
#pragma unroll
  for (int j = 0; j < 3; ++j) {
    int v = faces[3 * f + j];
    atomicAdd(&M[(size_t)v * NF + f], 1.0f);
  }
}

// ---------------------------------------------------------------------------
// Stage A: per-point geometry (points3d) + per-face feature accumulation.
// 128 threads per block, 128 points per block; all points in a block share
// one batch (NPB % 128 == 0). Lookup tables staged into LDS via async loads.
// local_feature is read once -> non-temporal to preserve L2 for faceSum/M.
// ---------------------------------------------------------------------------
__global__ void __launch_bounds__(128) k_points(
    const float* __restrict__ uv, const float* __restrict__ feat,
    const float* __restrict__ verts, const float* __restrict__ uv_verts,
    const int* __restrict__ faces, const int* __restrict__ face_inds,
    float* __restrict__ points3d, float* __restrict__ faceSum,
    float* __restrict__ faceCnt) {
  __shared__ int   s_faces[NF * 3];      // 15360 B
  __shared__ float s_uvv[NV * 2];        //  5136 B
  __shared__ float s_verts[NV * 3];      //  7704 B
  __shared__ int   s_fidx[128];          //   512 B

  const int t = threadIdx.x;
  // cooperative async staging of the small tables (ASYNCcnt-tracked)
  stage_to_lds(s_faces, faces, NF * 3, t, 128);
  stage_to_lds(s_uvv, uv_verts, NV * 2, t, 128);
  stage_to_lds(s_verts, verts, NV * 3, t, 128);
  wait_async_then_barrier();

  const size_t base = (size_t)blockIdx.x * 128;   // first point of this block
  const int i = (int)base + t;                    // my point
  const int b = i >> 15;                          // batch (NPB = 2^15)

  // ---- geometry phase -----------------------------------------------------
  float pu = __builtin_nontemporal_load(uv + 2 * i + 0);
  float pv = __builtin_nontemporal_load(uv + 2 * i + 1);
  int xi = (int)roundf(pu * (float)(UVW - 1));
  int yi = (int)roundf(pv * (float)(UVW - 1));
  int fidx = face_inds[yi * UVW + xi];             // reused table: keep cached

  int i0 = s_faces[3 * fidx + 0];
  int i1 = s_faces[3 * fidx + 1];
  int i2 = s_faces[3 * fidx + 2];

  float3 vA = uvTo3d(s_uvv[2 * i0], s_uvv[2 * i0 + 1]);
  float3 vB = uvTo3d(s_uvv[2 * i1], s_uvv[2 * i1 + 1]);
  float3 vC = uvTo3d(s_uvv[2 * i2], s_uvv[2 * i2 + 1]);
  float3 p3 = uvTo3d(pu, pv);

  float3 AB = sub3(vB, vA), AC = sub3(vC, vA), BC = sub3(vC, vB);
  float3 AP = sub3(p3, vA), BP = sub3(p3, vB);

  float areaBAC = norm3(cross3(AB, AC));
  float w  = norm3(cross3(AB, AP)) / areaBAC;
  float vv = norm3(cross3(AC, AP)) / areaBAC;
  float uu = norm3(cross3(BC, BP)) / areaBAC;
  float s = fmaxf(fabsf(uu) + fabsf(vv) + fabsf(w), 1e-12f);
  uu /= s; vv /= s; w /= s;

  float ox = uu * s_verts[3 * i0 + 0] + vv * s_verts[3 * i1 + 0] + w * s_verts[3 * i2 + 0];
  float oy = uu * s_verts[3 * i0 + 1] + vv * s_verts[3 * i1 + 1] + w * s_verts[3 * i2 + 1];
  float oz = uu * s_verts[3 * i0 + 2] + vv * s_verts[3 * i1 + 2] + w * s_verts[3 * i2 + 2];
  __builtin_nontemporal_store(ox, points3d + 3 * i + 0);   // write-once output
  __builtin_nontemporal_store(oy, points3d + 3 * i + 1);
  __builtin_nontemporal_store(oz, points3d + 3 * i + 2);

  atomicAdd(&faceCnt[b * NF + fidx], 1.0f);
  s_fidx[t] = fidx;
  __syncthreads();

  // ---- feature scatter phase: thread t owns channel t ---------------------
  const int bb = (int)(base >> 15);
  float* fs = faceSum + ((size_t)bb * NF) * CH + t;
  for (int p = 0; p < 128; ++p) {
    int f = s_fidx[p];
    // 256 MB streamed exactly once: non-temporal, coalesced 512B bursts
    float val = __builtin_nontemporal_load(feat + (base + (size_t)p) * CH + t);
    atomicAdd(&fs[(size_t)f * CH], val);
  }
}

// ---------------------------------------------------------------------------
// Per-vertex counts: cntV[b,v] = sum_f M[v,f] * faceCnt[b,f]   (exact ints)
// ---------------------------------------------------------------------------
__global__ void k_counts(const float* __restrict__ M,
                         const float* __restrict__ faceCnt,
                         float* __restrict__ cntV) {
  int idx = blockIdx.x * blockDim.x + threadIdx.x;
  if (idx >= 16 * NVP) return;
  int b = idx / NVP, v = idx % NVP;
  const float* Mr = M + (size_t)v * NF;
  const float* Fc = faceCnt + b * NF;
  float s = 0.0f;
  for (int f = 0; f < NF; ++f) s = fmaf(Mr[f], Fc[f], s);
  cntV[idx] = s;
}

// ---------------------------------------------------------------------------
// Stage B: vertSum = M (NVPxNF) * faceSum_b (NFxCH) via V_WMMA_F32_16X16X4_F32,
// fused mean division in the epilogue. One wave per 16x16 output tile.
// A 16x4 layout: VGPR0/1 hold K={0,1} (lanes 0-15) and K={2,3} (lanes 16-31).
// B 4x16 mirrors per-row; D uses the documented 8-VGPR C/D layout.
// ---------------------------------------------------------------------------
__global__ void __launch_bounds__(32) k_gemm(const float* __restrict__ M,
                                             const float* __restrict__ S,
                                             const float* __restrict__ cntV,
                                             float* __restrict__ out2) {
  const int mt = blockIdx.x;       // 0..40  vertex tile
  const int nt = blockIdx.y;       // 0..7   channel tile
  const int b  = blockIdx.z;       // 0..15  batch
  const int lane = threadIdx.x;
  const int r  = lane & 15;
  const int hk = (lane & 16) ? 2 : 0;

  const float* Mrow = M + (size_t)(mt * 16 + r) * NF;
  const float* Sp   = S + (size_t)b * NF * CH + nt * 16 + r;

  __builtin_prefetch(Mrow, 0, 1);
  __builtin_prefetch(Sp, 0, 1);

  v8f acc = {};
  for (int k = 0; k < NF; k += 4) {
    v2f a = *(const v2f*)(Mrow + k + hk);              // M[row, k+hk .. k+hk+1]
    v2f bm;
    bm.x = Sp[(size_t)(k + hk) * CH];                  // S[k+hk,   col]
    bm.y = Sp[(size_t)(k + hk + 1) * CH];              // S[k+hk+1, col]
    acc = __builtin_amdgcn_wmma_f32_16x16x4_f32(
        /*neg_a=*/false, a, /*neg_b=*/false, bm,
        /*c_mod=*/(short)0, acc, /*reuse_a=*/false, /*reuse_b=*/false);
  }

  const int colbase = nt * 16 + r;
  const int rowadd  = (lane & 16) ? 8 : 0;
#pragma unroll
  for (int i = 0; i < 8; ++i) {
    int vrow = mt * 16 + i + rowadd;
    if (vrow < NV) {
      float c = cntV[b * NVP + vrow];
      out2[((size_t)b * NV + vrow) * CH + colbase] = acc[i] / c;  // 0/0 -> NaN, matches ref
    }
  }
}

// ---------------------------------------------------------------------------
extern "C" void kernel_launch(void* const* d_in, const int* in_sizes, int n_in,
                              void* d_out, int out_size, void* d_ws, size_t ws_size,
                              hipStream_t stream) {
  (void)in_sizes; (void)n_in; (void)out_size; (void)ws_size;
  const float* uv        = (const float*)d_in[0];
  const float* feat      = (const float*)d_in[1];
  const float* verts     = (const float*)d_in[2];
  const float* uv_verts  = (const float*)d_in[3];
  const int*   faces     = (const int*)d_in[4];
  const int*   face_inds = (const int*)d_in[5];

  float* points3d = (float*)d_out;                        // BN*3 floats
  float* out2     = points3d + (size_t)BN * 3;            // 16*NV*CH floats

  char* ws = (char*)d_ws;
  const size_t szFaceSum = (size_t)16 * NF * CH * 4;      // 10,485,760
  const size_t szFaceCnt = (size_t)16 * NF * 4;           //     81,920
  const size_t szM       = (size_t)NVP * NF * 4;          //  3,358,720
  float* faceSum = (float*)ws;
  float* faceCnt = (float*)(ws + szFaceSum);
  float* Mmat    = (float*)(ws + szFaceSum + szFaceCnt);
  float* cntV    = (float*)(ws + szFaceSum + szFaceCnt + szM);

  // zero the accumulation buffers (capture-safe)
  (void)hipMemsetAsync(ws, 0, szFaceSum + szFaceCnt + szM, stream);

  k_build_M<<<(NF + 127) / 128, 128, 0, stream>>>(faces, Mmat);
  k_points<<<BN / 128, 128, 0, stream>>>(uv, feat, verts, uv_verts, faces,
                                         face_inds, points3d, faceSum, faceCnt);
  k_counts<<<(16 * NVP + 255) / 256, 256, 0, stream>>>(Mmat, faceCnt, cntV);
  dim3 g(NVP / 16, CH / 16, 16);
  k_gemm<<<g, 32, 0, stream>>>(Mmat, faceSum, cntV, out2);
}